// PoseCorrection_77180562309221
// MI455X (gfx1250) — compile-verified
//
#include <hip/hip_runtime.h>
#include <hip/hip_bf16.h>
#include <cstdint>

#define GLB_AS __attribute__((address_space(1)))
#define LDS_AS __attribute__((address_space(3)))

typedef __attribute__((ext_vector_type(4))) int   v4i;
typedef __attribute__((ext_vector_type(4))) float v4f;

constexpr int kBlock     = 256;   // 8 waves (wave32) per workgroup
constexpr int kMaxFrames = 1024;  // LDS = 1024*6*4 = 24 KB  (<< 320 KB/WGP)

__global__ __launch_bounds__(kBlock)
void PoseCorrection_se3_kernel(const float* __restrict__ rays,
                               const float* __restrict__ corr_dict,
                               const int*   __restrict__ image_indices,
                               const int*   __restrict__ depth_mask,
                               float*       __restrict__ out,
                               int N, int n_frames)
{
    __shared__ __align__(16) float s_corr[kMaxFrames * 6];

    // ---- Stage correction_dict (n_frames x 6 f32) into LDS ----------------
    // Regular-temporal: this 24 KB table is re-read by every block and we
    // WANT it resident in L2 (192 MB); the streams below are NT instead.
    const int nfloats = n_frames * 6;
    const int nvec4   = nfloats >> 2;           // 1500 for n_frames=1000
#if __has_builtin(__builtin_amdgcn_global_load_async_to_lds_b128) && \
    __has_builtin(__builtin_amdgcn_global_load_async_to_lds_b32)
    // gfx1250 async DMA path: global -> LDS without touching VGPRs (ASYNCcnt)
    for (int v = threadIdx.x; v < nvec4; v += kBlock) {
        __builtin_amdgcn_global_load_async_to_lds_b128(
            (GLB_AS v4i*)(uintptr_t)(corr_dict + 4 * v),
            (LDS_AS v4i*)(LDS_AS void*)(&s_corr[4 * v]),
            0, 0);
    }
    for (int v = (nvec4 << 2) + threadIdx.x; v < nfloats; v += kBlock) {
        __builtin_amdgcn_global_load_async_to_lds_b32(
            (GLB_AS int*)(uintptr_t)(corr_dict + v),
            (LDS_AS int*)(LDS_AS void*)(&s_corr[v]),
            0, 0);
    }
  #if __has_builtin(__builtin_amdgcn_s_wait_asynccnt)
    __builtin_amdgcn_s_wait_asynccnt(0);
  #else
    asm volatile("s_wait_asynccnt 0" ::: "memory");
  #endif
#else
    // Fallback: VGPR round-trip staging
    for (int v = threadIdx.x; v < nvec4; v += kBlock)
        reinterpret_cast<v4f*>(s_corr)[v] =
            reinterpret_cast<const v4f*>(corr_dict)[v];
    for (int v = (nvec4 << 2) + threadIdx.x; v < nfloats; v += kBlock)
        s_corr[v] = corr_dict[v];
#endif
    __syncthreads();

    // ---- Streaming main loop: one 32-byte ray per lane --------------------
    // All per-ray traffic is single-touch -> non-temporal (TH=NT) so the
    // 152 MB stream does not evict the L2-resident correction table.
    const int tid    = blockIdx.x * kBlock + threadIdx.x;
    const int stride = gridDim.x * kBlock;
    const v4f* __restrict__ rays4 = reinterpret_cast<const v4f*>(rays);
    v4f* __restrict__ out4        = reinterpret_cast<v4f*>(out);

    for (int i = tid; i < N; i += stride) {
        const v4f a = __builtin_nontemporal_load(&rays4[2 * i + 0]); // r0..r3
        const v4f b = __builtin_nontemporal_load(&rays4[2 * i + 1]); // r4..r7
        const int m = __builtin_nontemporal_load(&depth_mask[i]);
        int idx     = __builtin_nontemporal_load(&image_indices[i]);
        idx = (idx < 0) ? 0 : (idx >= n_frames ? n_frames - 1 : idx);

        float r0 = 0.f, r1 = 0.f, r2 = 0.f;   // rho
        float p0 = 0.f, p1 = 0.f, p2 = 0.f;   // phi
        if (m == 1) {
            const int j = idx * 6;
            r0 = s_corr[j + 0]; r1 = s_corr[j + 1]; r2 = s_corr[j + 2];
            p0 = s_corr[j + 3]; p1 = s_corr[j + 4]; p2 = s_corr[j + 5];
        }

        const float th2 = p0 * p0 + p1 * p1 + p2 * p2;
        const float th  = __builtin_amdgcn_sqrtf(th2);
        const bool  sm  = th < 1e-6f;
        const float ts  = sm ? 1.0f : th;
        float sn, cs;
        __sincosf(ts, &sn, &cs);
        const float its  = __builtin_amdgcn_rcpf(ts);
        const float its2 = its * its;
        const float A = sm ? 1.0f - th2 * (1.0f / 6.0f)  : sn * its;
        const float B = sm ? 0.5f - th2 * (1.0f / 24.0f) : (1.0f - cs) * its2;
        const float C = sm ? (1.0f / 6.0f) - th2 * (1.0f / 120.0f)
                           : (ts - sn) * its2 * its;

        // t = rho + B*(phi x rho) + C*(phi*(phi.rho) - th2*rho)
        const float pdr = p0 * r0 + p1 * r1 + p2 * r2;
        const float cr0 = p1 * r2 - p2 * r1;
        const float cr1 = p2 * r0 - p0 * r2;
        const float cr2 = p0 * r1 - p1 * r0;
        const float tx = r0 + B * cr0 + C * (p0 * pdr - th2 * r0);
        const float ty = r1 + B * cr1 + C * (p1 * pdr - th2 * r1);
        const float tz = r2 + B * cr2 + C * (p2 * pdr - th2 * r2);

        // rot_dir = d + A*(phi x d) + B*(phi*(phi.d) - th2*d)
        const float d0 = a.w, d1 = b.x, d2 = b.y;
        const float pdd = p0 * d0 + p1 * d1 + p2 * d2;
        const float cd0 = p1 * d2 - p2 * d1;
        const float cd1 = p2 * d0 - p0 * d2;
        const float cd2 = p0 * d1 - p1 * d0;
        const float q0 = d0 + A * cd0 + B * (p0 * pdd - th2 * d0);
        const float q1 = d1 + A * cd1 + B * (p1 * pdd - th2 * d1);
        const float q2 = d2 + A * cd2 + B * (p2 * pdd - th2 * d2);

        v4f o0, o1;
        o0.x = a.x + tx; o0.y = a.y + ty; o0.z = a.z + tz; o0.w = q0;
        o1.x = q1;       o1.y = q2;
        // reference quirk: ret.at[6:] indexes ROWS -> rows 0..5 zero cols 6,7
        const bool keep = (i >= 6);
        o1.z = keep ? b.z : 0.0f;
        o1.w = keep ? b.w : 0.0f;
        __builtin_nontemporal_store(o0, &out4[2 * i + 0]);
        __builtin_nontemporal_store(o1, &out4[2 * i + 1]);
    }
}

extern "C" void kernel_launch(void* const* d_in, const int* in_sizes, int n_in,
                              void* d_out, int out_size, void* d_ws, size_t ws_size,
                              hipStream_t stream) {
    (void)n_in; (void)out_size; (void)d_ws; (void)ws_size;
    const float* rays  = (const float*)d_in[0];
    const float* cdict = (const float*)d_in[1];
    const int*   idx   = (const int*)d_in[2];
    const int*   mask  = (const int*)d_in[3];
    float*       out   = (float*)d_out;

    const int N        = in_sizes[0] / 8;
    int       n_frames = in_sizes[1] / 6;
    if (n_frames > kMaxFrames) n_frames = kMaxFrames;

    int blocks = (N + kBlock - 1) / kBlock;
    if (blocks > 2048) blocks = 2048;   // persistent-ish: bounds LDS-fill traffic
    if (blocks < 1) blocks = 1;

    PoseCorrection_se3_kernel<<<blocks, kBlock, 0, stream>>>(
        rays, cdict, idx, mask, out, N, n_frames);
}